// __AddDecomposedRelativePositionsPT_74723841016423
// MI455X (gfx1250) — compile-verified
//
#include <hip/hip_runtime.h>

// CDNA5 / gfx1250 fused "add decomposed relative positions" kernel.
// Memory-bound: streams attn (512MB) -> out (512MB) with NT cache hints;
// rel_h computed with V_WMMA_F32_16X16X4_F32, rel_w (GEMV-shaped) with
// packed FMAs, both staged in LDS.

typedef __attribute__((ext_vector_type(2))) float v2f;
typedef __attribute__((ext_vector_type(4))) float v4f;
typedef __attribute__((ext_vector_type(8))) float v8f;

__global__ __launch_bounds__(256)
void relpos_fused_kernel(const float* __restrict__ attn,
                         const float* __restrict__ q,
                         const float* __restrict__ rph,
                         const float* __restrict__ rpw,
                         float* __restrict__ out)
{
    constexpr int QW = 64, KH = 64, KW = 64, C = 64;

    __shared__ float Qt[16][C];     // q tile: 16 w-rows x 64 channels (4KB)
    __shared__ float relH[16][KH];  // rel_h[w, kh]                     (4KB)
    __shared__ float relW[16][KW];  // rel_w[w, kw]                     (4KB)

    const int bh  = blockIdx.x;          // b*64 + h
    const int h   = bh & 63;
    const int w0  = blockIdx.y << 4;     // w quarter: 0,16,32,48
    const int tid = threadIdx.x;

    // ---- Stage 1: load q[b,h, w0..w0+15, 0..63] into LDS (one float4/thread)
    {
        const int wl = tid >> 4;
        const int c4 = (tid & 15) << 2;
        const float4 v = *reinterpret_cast<const float4*>(
            q + (((size_t)bh * QW + (w0 + wl)) * C + c4));
        *reinterpret_cast<float4*>(&Qt[wl][c4]) = v;
    }
    __syncthreads();

    const int wave = tid >> 5;   // 8 waves of 32 (wave32)
    const int lane = tid & 31;
    const int half = lane >> 4;  // 0: lanes 0-15, 1: lanes 16-31
    const int lm   = lane & 15;

    if (wave < 4) {
        // ---- Stage 2a: rel_h 16x16 tile per wave via WMMA f32 16x16x4.
        // D[w, k0+n] = sum_c Qt[w][c] * rph[(h - (k0+n) + 63)*C + c]
        const int k0 = wave << 4;
        const float* __restrict__ brow =
            rph + (size_t)(h - (k0 + lm) + 63) * C;   // row for this lane's N
        v8f acc = {};
        #pragma unroll
        for (int c0 = 0; c0 < C; c0 += 4) {
            // A (16x4 f32): lane holds A[lm, c0+2*half .. +1] -> contiguous pair
            v2f a, b;
            a.x = Qt[lm][c0 + 2 * half + 0];
            a.y = Qt[lm][c0 + 2 * half + 1];
            // B (4x16 f32): lane holds B[2*half .. +1, lm] = rph row, contiguous
            b.x = brow[c0 + 2 * half + 0];
            b.y = brow[c0 + 2 * half + 1];
            acc = __builtin_amdgcn_wmma_f32_16x16x4_f32(
                false, a, false, b, (short)0, acc, false, false);
        }
        // C/D layout: VGPR r -> M = r + 8*half, N = lm
        #pragma unroll
        for (int r = 0; r < 8; ++r)
            relH[r + 8 * half][k0 + lm] = acc[r];
    } else {
        // ---- Stage 2b: rel_w[wl, k] = sum_c Qt[wl][c] * rpw[(w0+wl-k+63)*C+c]
        // 16*64 = 1024 outputs across 128 threads (waves 4-7).
        for (int o = tid - 128; o < 16 * KW; o += 128) {
            const int wl = o >> 6;
            const int k  = o & 63;
            const float* __restrict__ trow =
                rpw + (size_t)(w0 + wl - k + 63) * C;
            float sx = 0.f, sy = 0.f, sz = 0.f, sw = 0.f;
            #pragma unroll
            for (int c = 0; c < C; c += 4) {
                const float4 qa = *reinterpret_cast<const float4*>(&Qt[wl][c]);
                const float4 tb = *reinterpret_cast<const float4*>(&trow[c]);
                sx = fmaf(qa.x, tb.x, sx);
                sy = fmaf(qa.y, tb.y, sy);
                sz = fmaf(qa.z, tb.z, sz);
                sw = fmaf(qa.w, tb.w, sw);
            }
            relW[wl][k] = (sx + sy) + (sz + sw);
        }
    }
    __syncthreads();

    // ---- Stage 3: stream 16 x 4096 floats: out = attn + relH + relW.
    // attn/out are single-touch 512MB streams (> 192MB L2): use non-temporal
    // b128 load/store (TH=NT) so L2 keeps the reused rel tables / q rows.
    const size_t base = ((size_t)bh * QW + w0) * (size_t)(KH * KW);
    #pragma unroll 4
    for (int i = tid; i < (16 * KH * KW) / 4; i += 256) {
        const int flat = i << 2;
        const int wl = flat >> 12;        // local w (0..15)
        const int p  = flat & 4095;       // position within (kh,kw) plane
        const int kh = p >> 6;
        const int kw = p & 63;            // multiple of 4
        const size_t off = base + (size_t)wl * (KH * KW) + p;

        v4f a = __builtin_nontemporal_load(
            reinterpret_cast<const v4f*>(attn + off));
        const float rh = relH[wl][kh];
        const float4 rw = *reinterpret_cast<const float4*>(&relW[wl][kw]);
        a.x = (a.x + rh) + rw.x;
        a.y = (a.y + rh) + rw.y;
        a.z = (a.z + rh) + rw.z;
        a.w = (a.w + rh) + rw.w;
        __builtin_nontemporal_store(a, reinterpret_cast<v4f*>(out + off));
    }
}

extern "C" void kernel_launch(void* const* d_in, const int* in_sizes, int n_in,
                              void* d_out, int out_size, void* d_ws, size_t ws_size,
                              hipStream_t stream) {
    (void)in_sizes; (void)n_in; (void)d_ws; (void)ws_size; (void)out_size;
    const float* attn = (const float*)d_in[0];   // (8, 4096, 4096)
    const float* q    = (const float*)d_in[1];   // (8, 4096, 64)
    const float* rph  = (const float*)d_in[2];   // (127, 64)
    const float* rpw  = (const float*)d_in[3];   // (127, 64)
    // d_in[4..7] = q_h,q_w,k_h,k_w (all 64; static in this problem)

    dim3 grid(8 * 64, 4);   // (b*qh, w-quarter)
    relpos_fused_kernel<<<grid, 256, 0, stream>>>(
        attn, q, rph, rpw, (float*)d_out);
}